// LayerCase_74174085202697
// MI455X (gfx1250) — compile-verified
//
#include <hip/hip_runtime.h>
#include <stdint.h>

// Problem constants from the reference
#define T_DIM 2048
#define N_DIM 64
#define C_DIM 256
#define S_DIM 256
#define E_DIM 513            // 2*S + 1
#define NEGV  (-1e30f)
#define NB    4              // x-row pipeline depth (power of two)

// ---------------------------------------------------------------------------
// CDNA5 async global -> LDS copy (GLOBAL_LOAD_ASYNC_TO_LDS_B32, ASYNCcnt).
// lds_addr = per-lane LDS byte offset; gsrc = per-lane 64-bit global address.
// ---------------------------------------------------------------------------
__device__ __forceinline__ void async_row_b32(uint32_t lds_addr, const float* gsrc) {
  asm volatile("global_load_async_to_lds_b32 %0, %1, off"
               :
               : "v"(lds_addr), "v"(gsrc)
               : "memory");
}

template <int CNT>
__device__ __forceinline__ void wait_async() {
#if __has_builtin(__builtin_amdgcn_s_wait_asynccnt)
  __builtin_amdgcn_s_wait_asynccnt(CNT);
#else
  asm volatile("s_wait_asynccnt %0" :: "i"(CNT) : "memory");
#endif
}

// ---------------------------------------------------------------------------
// One workgroup per batch element.  512 threads: thread s owns extended
// label s; thread 0 additionally owns s = 512.  Alpha double-buffered in
// LDS; x rows (1 KB each) streamed with async-LDS copies, NB rows deep so
// HBM latency overlaps ~3 scan steps (per-wave async loads complete in
// order, so s_wait_asynccnt <= NB-2 guarantees row t+1 has landed).
// ---------------------------------------------------------------------------
__global__ __launch_bounds__(512) void ctc_alpha_kernel(
    const float* __restrict__ x,       // (T, N, C) log-probs
    const int*   __restrict__ y,       // (N, S) labels (>=1)
    const int*   __restrict__ ilen,    // (N,)
    const int*   __restrict__ tlen,    // (N,)
    float*       __restrict__ loss_ws) // (N,) per-batch loss / target_len
{
  __shared__ float xrow[NB][C_DIM];      // pipelined x[t, n, :] rows
  __shared__ float abuf[2][E_DIM + 3];   // double-buffered alpha (padded)

  const int tid  = threadIdx.x;          // s = tid (0..511)
  const int n    = blockIdx.x;
  const int Tlim = min(T_DIM, ilen[n]);  // freeze => loop only to Tlim-1

  // ext[s] / skip[s] in registers (odd s -> real label, even s -> blank)
  int  ext_s  = 0;
  bool skip_s = false;
  if (tid & 1) {
    const int li  = (tid - 1) >> 1;
    ext_s         = y[n * S_DIM + li];
    const int em2 = (li >= 1) ? y[n * S_DIM + li - 1] : -1;
    skip_s        = (ext_s != 0) && (ext_s != em2);
  }
  const bool use_skip = skip_s && (tid >= 2);  // s==1 has no s-2 predecessor

  const uint32_t x_lds_base = (uint32_t)(uintptr_t)(&xrow[0][0]);
  const uint32_t lane_off   = (uint32_t)tid << 2;
  const size_t   row_stride = (size_t)N_DIM * C_DIM;   // floats between rows t
  const float*   gp         = x + (size_t)n * C_DIM + tid;  // row 0, this lane

  float* A = abuf[0];
  float* B = abuf[1];

  // ---- prologue: issue rows 0..NB-1, init alpha from row 0 ------------
  if (tid < C_DIM) {
    #pragma unroll
    for (int r = 0; r < NB; ++r)
      if (r < Tlim)
        async_row_b32(x_lds_base + ((uint32_t)r << 10) + lane_off,
                      gp + (size_t)r * row_stride);
  }
  wait_async<NB - 1>();   // row 0 landed (issuing waves)
  __syncthreads();        // row 0 visible to all waves

  float a0;
  if      (tid == 0) a0 = xrow[0][0];        // alpha0[0] = lp[0, n, blank]
  else if (tid == 1) a0 = xrow[0][ext_s];    // alpha0[1] = lp[0, n, ext[1]]
  else               a0 = NEGV;
  A[tid] = a0;
  if (tid == 0) A[E_DIM - 1] = NEGV;         // s = 512

  wait_async<NB - 2>();   // row 1 landed
  __syncthreads();        // A + row 1 visible

  // gpre -> row (t + NB - 1) issued at iteration t (starts at row NB)
  const float* gpre = gp + (size_t)NB * row_stride;

  // ---- sequential scan over time --------------------------------------
  for (int t = 1; t < Tlim; ++t) {
    const int cur = t & (NB - 1);

    // issue row t+NB-1 into slot (t-1)%NB (last read at iter t-1)
    if ((tid < C_DIM) && (t + NB - 1 < Tlim)) {
      const uint32_t slot = (uint32_t)((t - 1) & (NB - 1));
      async_row_b32(x_lds_base + (slot << 10) + lane_off, gpre);
    }
    gpre += row_stride;

    // lse3 over {alpha[s], alpha[s-1], alpha[s-2] if skip}
    const float a1 = A[tid];
    const float a2 = (tid >= 1) ? A[tid - 1] : NEGV;
    const float a3 = use_skip   ? A[tid - 2] : NEGV;
    float m = fmaxf(a1, fmaxf(a2, a3));
    float v = m + __logf(__expf(a1 - m) + __expf(a2 - m) + __expf(a3 - m));
    B[tid] = v + xrow[cur][ext_s];

    if (tid == 0) {  // extra element s = 512 (even -> blank, no skip)
      const float b1 = A[E_DIM - 1];
      const float b2 = A[E_DIM - 2];
      float mm = fmaxf(fmaxf(b1, b2), NEGV);
      float vv = mm + __logf(__expf(b1 - mm) + __expf(b2 - mm) + __expf(NEGV - mm));
      B[E_DIM - 1] = vv + xrow[cur][0];
    }

    wait_async<NB - 2>();  // row t+1 landed; up to NB-2 copies still in flight
    __syncthreads();       // B complete + async data visible to all waves
    float* tmp = A; A = B; B = tmp;
  }

  // ---- terminal log-likelihood ----------------------------------------
  if (tid == 0) {
    const int   L  = tlen[n];
    const float la = A[2 * L - 1];
    const float lb = A[2 * L];
    const float m  = fmaxf(la, lb);
    const float ll = m + __logf(__expf(la - m) + __expf(lb - m));
    loss_ws[n] = -ll / (float)max(L, 1);
  }
}

// mean over the batch -> d_out[0]
__global__ void ctc_reduce_kernel(const float* __restrict__ loss_ws,
                                  float* __restrict__ out) {
  if (threadIdx.x == 0) {
    float s = 0.0f;
    for (int i = 0; i < N_DIM; ++i) s += loss_ws[i];
    out[0] = s / (float)N_DIM;
  }
}

extern "C" void kernel_launch(void* const* d_in, const int* in_sizes, int n_in,
                              void* d_out, int out_size, void* d_ws, size_t ws_size,
                              hipStream_t stream) {
  const float* x  = (const float*)d_in[0];  // (T, N, C) float32
  const int*   y  = (const int*)  d_in[1];  // (N, S)    int32
  const int*   il = (const int*)  d_in[2];  // (N,)      int32
  const int*   tl = (const int*)  d_in[3];  // (N,)      int32
  float* ws = (float*)d_ws;                 // N floats of scratch

  ctc_alpha_kernel<<<N_DIM, 512, 0, stream>>>(x, y, il, tl, ws);
  ctc_reduce_kernel<<<1, 32, 0, stream>>>(ws, (float*)d_out);
}